// BuildVolume2dChaos_5669356834311
// MI455X (gfx1250) — compile-verified
//
#include <hip/hip_runtime.h>
#include <cstdint>

// Problem constants from the reference: B=2, C=32, H=128, W=256, D=32, fp32.
namespace {
constexpr int kB = 2, kC = 32, kH = 128, kW = 256, kD = 32;
constexpr int kThreads   = 512;              // 16 wave32s per workgroup
constexpr int kXStride   = 36;               // floats per x column: 144 B (16B-aligned, bank-spread)
constexpr int kRowFloats = kW * kXStride;    // 9216 floats per staged row
constexpr int kHW        = kH * kW;          // 32768
}  // namespace

// CDNA5 async DMA: global -> LDS, 4 bytes per lane, tracked by ASYNCcnt.
__device__ __forceinline__ void async_load_b32_to_lds(uint32_t lds_byte_off,
                                                      const void* gaddr) {
  asm volatile("global_load_async_to_lds_b32 %0, %1, off"
               : : "v"(lds_byte_off), "v"(gaddr) : "memory");
}

__device__ __forceinline__ void wait_async_zero() {
  asm volatile("s_wait_asynccnt 0" : : : "memory");
}

__global__ __launch_bounds__(kThreads)
void build_volume_l1(const float* __restrict__ refimg,
                     const float* __restrict__ tgtimg,
                     const float* __restrict__ disps,
                     float* __restrict__ out) {
  // tgt row y0 | tgt row y1 | ref row h, each [x][c] channel-contiguous, padded.
  __shared__ float smem[3 * kRowFloats];     // 110592 B < 320 KB/WGP

  const int bh  = blockIdx.x;                // 0 .. B*H-1
  const int b   = bh >> 7;                   // kH == 128
  const int h   = bh & (kH - 1);
  const int tid = threadIdx.x;

  // Vertical resample position: iy = h*H/(H-1) - 0.5 (grid_sample align_corners=False
  // applied to an align_corners-style grid), identical for every pixel of this row.
  const float iy  = (float)h * (128.0f / 127.0f) - 0.5f;
  const float y0f = floorf(iy);
  const float wy1 = iy - y0f;
  const float wy0 = 1.0f - wy1;
  const int   y0  = (int)y0f;
  const int   y1  = y0 + 1;
  const float vy0 = (y0 >= 0 && y0 < kH) ? 1.0f : 0.0f;
  const float vy1 = (y1 >= 0 && y1 < kH) ? 1.0f : 0.0f;
  const int   y0c = min(max(y0, 0), kH - 1);
  const int   y1c = min(max(y1, 0), kH - 1);

  // Low 32 bits of a flat LDS pointer are the LDS byte offset (shared aperture).
  const uint32_t ldsBase = (uint32_t)(uintptr_t)(void*)smem;

  // All flat element indices fit in 31 bits (max ~2.1M): keep index math 32-bit.
  const int dhBase = ((b * kD) * kH + h) * kW;         // disps/out: + d*kHW + w

  // Warm L2 for this block's 32 KB disparity slab (exactly 512 distinct 64 B
  // lines: d = tid>>4, line offset (tid&15)*16 floats — always in range).
  __builtin_prefetch(&disps[dhBase + (tid >> 4) * kHW + (tid & 15) * 16], 0, 3);

  // ---- Stage 3 feature rows via async global->LDS DMA, transposed to [x][c] ----
  const float* gRow[3];
  gRow[0] = tgtimg + (size_t)((b * kC * kH + y0c) * kW);
  gRow[1] = tgtimg + (size_t)((b * kC * kH + y1c) * kW);
  gRow[2] = refimg + (size_t)((b * kC * kH + h)   * kW);

  #pragma unroll
  for (int r = 0; r < 3; ++r) {
    const uint32_t lb = ldsBase + (uint32_t)(r * kRowFloats) * 4u;
    for (int e = tid; e < kC * kW; e += kThreads) {   // 16 iters: lanes walk x => coalesced
      const int c = e >> 8;                           // kW == 256
      const int x = e & (kW - 1);
      async_load_b32_to_lds(lb + (uint32_t)(x * kXStride + c) * 4u,
                            gRow[r] + (c * kHW + x));
    }
  }
  wait_async_zero();     // each wave drains its own ASYNCcnt (48 outstanding max)
  __syncthreads();       // make DMA'd LDS visible across all 16 waves

  const float* row0 = smem;
  const float* row1 = smem + kRowFloats;
  const float* rrow = smem + 2 * kRowFloats;
  const float xScale = 256.0f / 255.0f;               // W/(W-1)

  // ---- Compute D*W = 8192 outputs per workgroup (16 per thread) ----
  #pragma unroll 2
  for (int i = tid; i < kD * kW; i += kThreads) {
    const int d = i >> 8;
    const int w = i & (kW - 1);
    const int gidx = dhBase + d * kHW + w;            // 32-bit flat index
    const float dv  = disps[gidx];                    // coalesced in w
    const float ix  = ((float)w - dv) * xScale - 0.5f;
    const float x0f = floorf(ix);
    const float wx1 = ix - x0f;
    const float wx0 = 1.0f - wx1;
    const int   x0  = (int)x0f;
    const int   x1  = x0 + 1;
    const float vx0 = (x0 >= 0 && x0 < kW) ? 1.0f : 0.0f;
    const float vx1 = (x1 >= 0 && x1 < kW) ? 1.0f : 0.0f;
    const int   x0c = min(max(x0, 0), kW - 1);
    const int   x1c = min(max(x1, 0), kW - 1);

    // zeros-padding: out-of-bounds taps contribute 0 via their weight
    const float w00 = wy0 * wx0 * vy0 * vx0;
    const float w01 = wy0 * wx1 * vy0 * vx1;
    const float w10 = wy1 * wx0 * vy1 * vx0;
    const float w11 = wy1 * wx1 * vy1 * vx1;

    const float* p00 = row0 + x0c * kXStride;
    const float* p01 = row0 + x1c * kXStride;
    const float* p10 = row1 + x0c * kXStride;
    const float* p11 = row1 + x1c * kXStride;
    const float* pr  = rrow + w   * kXStride;

    float sum = 0.0f;
    #pragma unroll
    for (int c = 0; c < kC; c += 4) {                 // 5x ds_load_b128 per iter
      const float4 a = *(const float4*)(p00 + c);
      const float4 e = *(const float4*)(p01 + c);
      const float4 f = *(const float4*)(p10 + c);
      const float4 g = *(const float4*)(p11 + c);
      const float4 r = *(const float4*)(pr  + c);
      sum += fabsf(r.x - (w00 * a.x + w01 * e.x + w10 * f.x + w11 * g.x));
      sum += fabsf(r.y - (w00 * a.y + w01 * e.y + w10 * f.y + w11 * g.y));
      sum += fabsf(r.z - (w00 * a.z + w01 * e.z + w10 * f.z + w11 * g.z));
      sum += fabsf(r.w - (w00 * a.w + w01 * e.w + w10 * f.w + w11 * g.w));
    }
    out[gidx] = sum;                                  // coalesced in w
  }
}

extern "C" void kernel_launch(void* const* d_in, const int* in_sizes, int n_in,
                              void* d_out, int out_size, void* d_ws, size_t ws_size,
                              hipStream_t stream) {
  (void)in_sizes; (void)n_in; (void)d_ws; (void)ws_size; (void)out_size;
  const float* refimg = (const float*)d_in[0];   // [B,C,H,W] fp32
  const float* tgtimg = (const float*)d_in[1];   // [B,C,H,W] fp32
  const float* disps  = (const float*)d_in[2];   // [B,D,H,W] fp32
  float* out = (float*)d_out;                    // [B,D,H,W] fp32

  dim3 grid(kB * kH);        // one workgroup per (batch, output row)
  dim3 block(kThreads);
  build_volume_l1<<<grid, block, 0, stream>>>(refimg, tgtimg, disps, out);
}